// MaxSimScores_87170656239782
// MI455X (gfx1250) — compile-verified
//
#include <hip/hip_runtime.h>

// MaxSim scores: B=16, N=100, T=256, D=128 (fp32 in, fp32 out).
// One workgroup per (b,n). LDS holds ONE 256x128 tile (time-multiplexed:
// Q first -> A-frags pulled into VGPRs -> overwritten by Doc tile), so
// ~143 KB LDS/WG -> 2 WGs per 320KB WGP (load/compute overlap).
// Wave w owns tq rows [32w,32w+32): A fragments register-resident,
// inner loop = 2 ds_load_b64 -> 4 v_wmma_f32_16x16x4_f32.
// Async HBM->LDS staging via global_load_async_to_lds_b128 when available.

#define BB 16
#define NN 100
#define TT 256
#define DD 128
#define STRIDE 132   // padded row stride in floats (float4-aligned staging)

typedef __attribute__((ext_vector_type(2))) float v2f;
typedef __attribute__((ext_vector_type(8))) float v8f;
typedef __attribute__((ext_vector_type(4))) int   v4i;

#if defined(__gfx1250__) && __has_builtin(__builtin_amdgcn_global_load_async_to_lds_b128)
#define USE_ASYNC 1
#else
#define USE_ASYNC 0
#endif

__device__ __forceinline__ void stage_tile(const float* __restrict__ g,
                                           float* s, int tid) {
    #pragma unroll
    for (int it = 0; it < (TT * DD / 4) / 256; ++it) {
        int i   = tid + it * 256;            // float4 index
        int row = i >> 5;                    // 32 float4 per row
        int c4  = (i & 31) << 2;             // float column
#if USE_ASYNC
        __builtin_amdgcn_global_load_async_to_lds_b128(
            (__attribute__((address_space(1))) v4i*)(g + (size_t)i * 4),
            (__attribute__((address_space(3))) v4i*)(s + row * STRIDE + c4),
            0, 0);
#else
        *(float4*)(s + row * STRIDE + c4) = *(const float4*)(g + (size_t)i * 4);
#endif
    }
#if USE_ASYNC
    asm volatile("s_wait_asynccnt 0x0" ::: "memory");
#endif
}

__device__ __forceinline__ void normalize_rows(float* s, int tid) {
    // thread t owns row t; rotated access -> conflict-free banks
    const int base = tid * STRIDE;
    float ss = 0.f;
    #pragma unroll 8
    for (int dd = 0; dd < DD; ++dd) {
        int d = (dd + tid) & (DD - 1);
        float x = s[base + d];
        ss += x * x;
    }
    float r = 1.0f / fmaxf(sqrtf(ss), 1e-12f);
    #pragma unroll 8
    for (int dd = 0; dd < DD; ++dd) {
        int d = (dd + tid) & (DD - 1);
        s[base + d] *= r;
    }
}

__global__ __launch_bounds__(256) void maxsim_kernel(const float* __restrict__ q,
                                                     const float* __restrict__ docs,
                                                     float* __restrict__ out) {
    __shared__ __align__(16) float buf[TT * STRIDE];   // Q tile, then Doc tile
    __shared__ float wavecol[8][TT];                   // per-wave column maxes
    __shared__ float partial[8];

    const int bn  = blockIdx.x;
    const int b   = bn / NN;
    const int n   = bn % NN;
    const int tid = threadIdx.x;

    const int lane    = tid & 31;
    const int wave    = tid >> 5;
    const int lrow    = lane & 15;
    const int halfsel = lane >> 4;          // 0: K pair (0,1); 1: K pair (2,3)

    // ---- Phase 1: Q[b] -> LDS, normalize, pull A-frags into VGPRs ----
    stage_tile(q + (size_t)b * TT * DD, buf, tid);
    __syncthreads();
    normalize_rows(buf, tid);
    __syncthreads();

    v2f aQ0[32], aQ1[32];   // wave's 2 tq blocks, all 32 K-steps (128 VGPRs)
    {
        const float* q0 = &buf[(wave * 32 + lrow) * STRIDE + 2 * halfsel];
        const float* q1 = q0 + 16 * STRIDE;
        #pragma unroll
        for (int j = 0; j < 32; ++j) {
            aQ0[j] = *(const v2f*)(q0 + j * 4);
            aQ1[j] = *(const v2f*)(q1 + j * 4);
        }
    }
    __syncthreads();   // all reads of Q done before overwrite

    // ---- Phase 2: Doc[b,n] -> LDS (same buffer), normalize ----
    stage_tile(docs + ((size_t)b * NN + n) * (size_t)TT * DD, buf, tid);
    __syncthreads();
    normalize_rows(buf, tid);
    __syncthreads();

    // ---- Phase 3: GEMM. All td tiles, this wave's 2 tq blocks. ----
    for (int tdb = 0; tdb < 16; tdb += 2) {
        const float* b0p = &buf[(tdb * 16 + lrow) * STRIDE + 2 * halfsel];
        const float* b1p = b0p + 16 * STRIDE;

        v8f c00 = {0.f,0.f,0.f,0.f,0.f,0.f,0.f,0.f};
        v8f c10 = {0.f,0.f,0.f,0.f,0.f,0.f,0.f,0.f};
        v8f c01 = {0.f,0.f,0.f,0.f,0.f,0.f,0.f,0.f};
        v8f c11 = {0.f,0.f,0.f,0.f,0.f,0.f,0.f,0.f};

        #pragma unroll
        for (int j = 0; j < 32; ++j) {
            v2f b0 = *(const v2f*)(b0p + j * 4);   // B frag, td tile tdb
            v2f b1 = *(const v2f*)(b1p + j * 4);   // B frag, td tile tdb+1
            c00 = __builtin_amdgcn_wmma_f32_16x16x4_f32(false, aQ0[j], false, b0,
                                                        (short)0, c00, false, false);
            c10 = __builtin_amdgcn_wmma_f32_16x16x4_f32(false, aQ1[j], false, b0,
                                                        (short)0, c10, false, false);
            c01 = __builtin_amdgcn_wmma_f32_16x16x4_f32(false, aQ0[j], false, b1,
                                                        (short)0, c01, false, false);
            c11 = __builtin_amdgcn_wmma_f32_16x16x4_f32(false, aQ1[j], false, b1,
                                                        (short)0, c11, false, false);
        }

        // Column max over this wave's 32 tq rows (both tq blocks)
        float m0 = fmaxf(c00[0], c10[0]);
        float m1 = fmaxf(c01[0], c11[0]);
        #pragma unroll
        for (int j2 = 1; j2 < 8; ++j2) {
            m0 = fmaxf(m0, fmaxf(c00[j2], c10[j2]));
            m1 = fmaxf(m1, fmaxf(c01[j2], c11[j2]));
        }
        m0 = fmaxf(m0, __shfl_xor(m0, 16, 32));   // merge row halves (same td col)
        m1 = fmaxf(m1, __shfl_xor(m1, 16, 32));
        if (lane < 16) {
            wavecol[wave][tdb * 16 + lrow]      = m0;
            wavecol[wave][tdb * 16 + 16 + lrow] = m1;
        }
    }
    __syncthreads();

    // ---- Phase 4: max across waves (full 256 tq rows), mean over td ----
    float m = wavecol[0][tid];
    #pragma unroll
    for (int w = 1; w < 8; ++w) m = fmaxf(m, wavecol[w][tid]);
    #pragma unroll
    for (int off = 16; off >= 1; off >>= 1)
        m += __shfl_xor(m, off, 32);
    if (lane == 0) partial[wave] = m;
    __syncthreads();

    if (tid == 0) {
        float s = 0.f;
        #pragma unroll
        for (int w = 0; w < 8; ++w) s += partial[w];
        out[bn] = s * (1.0f / (float)TT);   // mean over 256 doc tokens
    }
}

extern "C" void kernel_launch(void* const* d_in, const int* in_sizes, int n_in,
                              void* d_out, int out_size, void* d_ws, size_t ws_size,
                              hipStream_t stream) {
    const float* q    = (const float*)d_in[0];   // (16, 256, 128) fp32
    const float* docs = (const float*)d_in[1];   // (16, 100, 256, 128) fp32
    float* out = (float*)d_out;                  // (16, 100) fp32
    (void)in_sizes; (void)n_in; (void)out_size; (void)d_ws; (void)ws_size;

    dim3 grid(BB * NN);
    dim3 block(256);
    maxsim_kernel<<<grid, block, 0, stream>>>(q, docs, out);
}